// Model_26353919328451
// MI455X (gfx1250) — compile-verified
//
#include <hip/hip_runtime.h>
#include <hip/hip_bf16.h>
#include <math.h>

// ---------------------------------------------------------------------------
// Fused 2-layer RNN (T=2048, B=2048, EMB=10, HID=8, NCLS=4) for gfx1250.
// One wave (32 threads) owns a tile of 16 batch sequences and walks all
// T steps; state + all weights stay in VGPRs. All matmuls are
// V_WMMA_F32_16X16X4_F32 chains in [hidden(M) x batch(N)] orientation.
//
// Row-duplication trick: every weight A-operand repeats rows 0-7 into rows
// 8-15, so the WMMA D output holds h[v][lo] in *both* half-waves. The next
// B-operand is then built with 4 in-lane v_cndmask ops - no shuffles, no LDS.
// ---------------------------------------------------------------------------

#define RNN_T    2048
#define RNN_B    2048
#define RNN_EMB  10
#define RNN_HID  8
#define RNN_NCLS 4

typedef __attribute__((ext_vector_type(2))) float v2f;
typedef __attribute__((ext_vector_type(4))) float v4f;
typedef __attribute__((ext_vector_type(8))) float v8f;

#if defined(__has_builtin)
#if __has_builtin(__builtin_amdgcn_tanhf)
#define RNN_TANH(x) __builtin_amdgcn_tanhf(x)
#endif
#endif
#ifndef RNN_TANH
#define RNN_TANH(x) tanhf(x)
#endif

// D = A(16x4 f32) * B(4x16 f32) + C(16x16 f32)
__device__ __forceinline__ v8f wmma4(v2f a, v2f b, v8f c) {
    return __builtin_amdgcn_wmma_f32_16x16x4_f32(
        /*neg_a=*/false, a, /*neg_b=*/false, b,
        /*c_mod=*/(short)0, c, /*reuse_a=*/false, /*reuse_b=*/false);
}

// A-operand chunk (K = 4c..4c+3) of an [8 x 8] row-major weight matrix with
// rows 8..15 duplicating rows 0..7 (M = lo & 7).
// Per-lane: V0 -> k = 4c + 2*hi, V1 -> k = 4c + 2*hi + 1.
__device__ __forceinline__ v2f load_a_dup(const float* W, int lo, int hi, int c) {
    const int r  = lo & 7;
    const int k0 = 4 * c + 2 * hi;
    v2f a;
    a.x = W[r * 8 + k0];
    a.y = W[r * 8 + k0 + 1];
    return a;
}

// Same but only `valid` leading rows are non-zero (for W_fc: 4 classes).
__device__ __forceinline__ v2f load_a_fc(const float* W, int lo, int hi, int c) {
    const int r = lo & 7;
    v2f a; a.x = 0.0f; a.y = 0.0f;
    if (r < RNN_NCLS) {
        const int k0 = 4 * c + 2 * hi;
        a.x = W[r * 8 + k0];
        a.y = W[r * 8 + k0 + 1];
    }
    return a;
}

// hD[v] holds h[v][lo] in every lane (thanks to row duplication), so the
// B-operand chunks are pure in-lane selects on the half-wave index.
__device__ __forceinline__ void d_to_b(const v8f& hD, bool hi, v2f& b0, v2f& b1) {
    b0.x = hi ? hD[2] : hD[0];
    b0.y = hi ? hD[3] : hD[1];
    b1.x = hi ? hD[6] : hD[4];
    b1.y = hi ? hD[7] : hD[5];
}

__global__ __launch_bounds__(32)
void rnn_fused_kernel(const int* __restrict__ x,
                      const float* __restrict__ emb,
                      const float* __restrict__ W_ih0,
                      const float* __restrict__ W_hh0,
                      const float* __restrict__ b_ih0,
                      const float* __restrict__ b_hh0,
                      const float* __restrict__ W_ih1,
                      const float* __restrict__ W_hh1,
                      const float* __restrict__ b_ih1,
                      const float* __restrict__ b_hh1,
                      const float* __restrict__ W_fc,
                      const float* __restrict__ b_fc,
                      float* __restrict__ out) {
    const int lane = threadIdx.x & 31;
    const int lo   = lane & 15;        // batch column within tile (N index)
    const int hi   = lane >> 4;        // half-wave select
    const int batch_base = blockIdx.x * 16;

    // ---- table0^T as an A operand (M=j hidden row, K=v vocab, K=4 exact) --
    // table0[v][j] = sum_d emb[v,d]*W_ih0[j,d] + b_ih0[j] + b_hh0[j]
    v2f t0a;
    {
        const int j = lo & 7;                     // duplicated rows 8..15
        const float bias = b_ih0[j] + b_hh0[j];
        const int v0 = 2 * hi, v1 = 2 * hi + 1;
        float a0 = bias, a1 = bias;
        #pragma unroll
        for (int d = 0; d < RNN_EMB; ++d) {
            const float w = W_ih0[j * RNN_EMB + d];
            a0 += emb[v0 * RNN_EMB + d] * w;
            a1 += emb[v1 * RNN_EMB + d] * w;
        }
        t0a.x = a0; t0a.y = a1;
    }

    // ---- weight A operands (rows duplicated across halves) -----------------
    const v2f whh0_a0 = load_a_dup(W_hh0, lo, hi, 0);
    const v2f whh0_a1 = load_a_dup(W_hh0, lo, hi, 1);
    const v2f wih1_a0 = load_a_dup(W_ih1, lo, hi, 0);
    const v2f wih1_a1 = load_a_dup(W_ih1, lo, hi, 1);
    const v2f whh1_a0 = load_a_dup(W_hh1, lo, hi, 0);
    const v2f whh1_a1 = load_a_dup(W_hh1, lo, hi, 1);
    const v2f wfc_a0  = load_a_fc (W_fc,  lo, hi, 0);
    const v2f wfc_a1  = load_a_fc (W_fc,  lo, hi, 1);

    // ---- bias accumulator inits in C/D layout (duplicated halves) ----------
    v8f c_bias1, c_fc, c_zero;
    #pragma unroll
    for (int i = 0; i < 8; ++i) {
        c_bias1[i] = b_ih1[i] + b_hh1[i];                 // rows j (both halves)
        c_fc[i]    = (i < RNN_NCLS) ? b_fc[i] : 0.0f;
        c_zero[i]  = 0.0f;
    }

    // ---- state -------------------------------------------------------------
    v8f h0D = c_zero, h1D = c_zero;
    v2f h0b0 = {0.0f, 0.0f}, h0b1 = {0.0f, 0.0f};
    v2f h1b0 = {0.0f, 0.0f}, h1b1 = {0.0f, 0.0f};

    const int* xrow = x + (size_t)batch_base + lo;
    float*     op   = out + ((size_t)batch_base + lo) * RNN_NCLS;

    int        xi      = xrow[0];          // token for t=0
    const int* xp_next = xrow + RNN_B;     // next row to preload

    #pragma unroll 1
    for (int t = 0; t < RNN_T; ++t) {
        // Cover HBM latency: warm the line ~8 steps ahead, preload next token.
        if (t + 8 < RNN_T)
            __builtin_prefetch(xp_next + (size_t)7 * RNN_B, 0, 3);
        int xi_next = xi;
        if (t + 1 < RNN_T) xi_next = xp_next[0];

        // one-hot B operand from token id (K=v vocab)
        v2f oh;
        oh.x = (xi == 2 * hi)     ? 1.0f : 0.0f;
        oh.y = (xi == 2 * hi + 1) ? 1.0f : 0.0f;

        // layer 0: pre0 = table0^T * onehot + W_hh0 * h0_{t-1}
        v8f p0 = wmma4(t0a, oh, c_zero);
        p0 = wmma4(whh0_a0, h0b0, p0);
        p0 = wmma4(whh0_a1, h0b1, p0);
        #pragma unroll
        for (int i = 0; i < 8; ++i) h0D[i] = RNN_TANH(p0[i]);
        d_to_b(h0D, hi != 0, h0b0, h0b1);

        // layer 1: pre1 = bias1 + W_ih1 * h0_t + W_hh1 * h1_{t-1}
        v8f p1 = c_bias1;
        p1 = wmma4(wih1_a0, h0b0, p1);
        p1 = wmma4(wih1_a1, h0b1, p1);
        p1 = wmma4(whh1_a0, h1b0, p1);
        p1 = wmma4(whh1_a1, h1b1, p1);
        #pragma unroll
        for (int i = 0; i < 8; ++i) h1D[i] = RNN_TANH(p1[i]);
        d_to_b(h1D, hi != 0, h1b0, h1b1);

        // fc: logits = b_fc + W_fc * h1_t  (class c in VGPR index, batch in lane)
        v8f lg = c_fc;
        lg = wmma4(wfc_a0, h1b0, lg);
        lg = wmma4(wfc_a1, h1b1, lg);

        if (!hi) {
            v4f res; res.x = lg[0]; res.y = lg[1]; res.z = lg[2]; res.w = lg[3];
            *(v4f*)op = res;                      // global_store_b128, coalesced
        }

        xi = xi_next;
        xp_next += RNN_B;
        op += (size_t)RNN_B * RNN_NCLS;
    }
}

extern "C" void kernel_launch(void* const* d_in, const int* in_sizes, int n_in,
                              void* d_out, int out_size, void* d_ws, size_t ws_size,
                              hipStream_t stream) {
    (void)in_sizes; (void)n_in; (void)d_ws; (void)ws_size; (void)out_size;
    const int*   x     = (const int*)  d_in[0];
    const float* emb   = (const float*)d_in[1];
    const float* W_ih0 = (const float*)d_in[2];
    const float* W_hh0 = (const float*)d_in[3];
    const float* b_ih0 = (const float*)d_in[4];
    const float* b_hh0 = (const float*)d_in[5];
    const float* W_ih1 = (const float*)d_in[6];
    const float* W_hh1 = (const float*)d_in[7];
    const float* b_ih1 = (const float*)d_in[8];
    const float* b_hh1 = (const float*)d_in[9];
    const float* W_fc  = (const float*)d_in[10];
    const float* b_fc  = (const float*)d_in[11];
    float* out = (float*)d_out;

    dim3 grid(RNN_B / 16);   // 128 independent 16-batch tiles
    dim3 block(32);          // one wave per tile
    rnn_fused_kernel<<<grid, block, 0, stream>>>(
        x, emb, W_ih0, W_hh0, b_ih0, b_hh0,
        W_ih1, W_hh1, b_ih1, b_hh1, W_fc, b_fc, out);
}